// Global_GConv_85839216378272
// MI455X (gfx1250) — compile-verified
//
#include <hip/hip_runtime.h>
#include <stdint.h>

// Problem constants (from reference)
#define N_NODES 81924
#define EDGES   7
#define CIN     128
#define COUT    128
#define BATCHN  4
#define M_TOT   (BATCHN * N_NODES)   // 327696
#define M_TILE  128
#define KCH     64                    // K chunk (half an edge's 128 features)
#define NCH     (EDGES * 2)           // 14 chunks
#define APAD    68                    // LDS row stride in floats (bank-conflict-free: 68 % 64 == 4)

typedef float v2f __attribute__((ext_vector_type(2)));
typedef float v8f __attribute__((ext_vector_type(8)));

// CDNA5 async global->LDS copy, 16B per lane, tracked with ASYNCcnt.
__device__ __forceinline__ void async_b128(uint32_t lds_off, uint64_t gaddr) {
    asm volatile("global_load_async_to_lds_b128 %0, %1, off"
                 :: "v"(lds_off), "v"(gaddr)
                 : "memory");
}

__global__ void zero_ws_kernel(float* ws) {
    if (threadIdx.x < 64) ws[threadIdx.x] = 0.0f;   // 256B zero page for pad rows
}

__global__ __launch_bounds__(256)
void gconv_wmma_kernel(const float* __restrict__ x,
                       const float* __restrict__ W,
                       const float* __restrict__ bias,
                       const int*   __restrict__ neigh,
                       const float* __restrict__ zpage,
                       float* __restrict__ out)
{
    // Double-buffered A (gathered x rows) and B (W slice), + gather index table.
    __shared__ __align__(16) float Ab[2][M_TILE][APAD];
    __shared__ __align__(16) float Wb[2][COUT][APAD];
    __shared__ int idx_sh[M_TILE * EDGES];

    const int t  = threadIdx.x;
    const int m0 = blockIdx.x * M_TILE;

    // ---- Precompute gather source rows (replicates the reference's padded
    //      x_flat + (+i*NUM_NODES) indexing; pad row -> -1 -> zero page) ----
    for (int u = t; u < M_TILE * EDGES; u += 256) {
        int r = u / EDGES;
        int e = u - r * EDGES;
        int m = m0 + r;
        int src = -1;
        if (m < M_TOT) {
            unsigned mm = (unsigned)m;
            unsigned bi = mm / N_NODES;
            unsigned nn = mm - bi * N_NODES;
            int v = neigh[nn * EDGES + e];
            unsigned j  = (unsigned)v + bi * N_NODES;       // index into padded x_flat
            unsigned q  = j / (N_NODES + 1);
            unsigned rr = j - q * (N_NODES + 1);
            src = (rr == (unsigned)N_NODES) ? -1 : (int)(q * N_NODES + rr);
        }
        idx_sh[u] = src;
    }
    __syncthreads();

    const int tr = t >> 4;          // 0..15 : row group within a load pass
    const int tc = t & 15;          // 0..15 : 16B column within 64-float chunk
    const uint64_t xbase = (uint64_t)x;
    const uint64_t wbase = (uint64_t)W;
    const uint64_t zbase = (uint64_t)zpage + (uint32_t)(tc * 16);

    // Issue one chunk's async loads: 8 passes x (A + W) = 16 instrs per wave.
    auto issue_chunk = [&](int c) {
        const int e = c >> 1, h = c & 1, buf = c & 1;
        uint32_t lA = (uint32_t)(uint64_t)&Ab[buf][tr][tc * 4];
        uint32_t lW = (uint32_t)(uint64_t)&Wb[buf][tr][tc * 4];
        const int kb = h * 256 + tc * 16;                 // byte offset within the edge's 512B row
#pragma unroll
        for (int p = 0; p < 8; ++p) {
            const int row = p * 16 + tr;
            const int src = idx_sh[row * EDGES + e];
            uint64_t ga = (src >= 0) ? (xbase + (uint32_t)(src * (CIN * 4) + kb)) : zbase;
            async_b128(lA + (uint32_t)(p * 16 * APAD * 4), ga);
            uint64_t gw = wbase + (uint32_t)(row * (EDGES * CIN * 4) + e * (CIN * 4) + kb);
            async_b128(lW + (uint32_t)(p * 16 * APAD * 4), gw);
        }
    };

    // ---- Wave/lane mapping: 8 waves as 4(M) x 2(N); wave owns 32 rows x 64 cols ----
    const int lane = t & 31;
    const int w    = t >> 5;
    const int wm   = w & 3;                  // 0..3
    const int wn   = w >> 2;                 // 0..1
    const int l15  = lane & 15;
    const int khi  = (lane >> 4) << 1;       // K phase: lanes 16-31 hold K+2,K+3

    v8f acc[2][4] = {};

    issue_chunk(0);
    issue_chunk(1);

    for (int c = 0; c < NCH; ++c) {
        // Wait for this chunk's own async loads (16 remain = the prefetched chunk).
        if (c + 1 < NCH) asm volatile("s_wait_asynccnt 16" ::: "memory");
        else             asm volatile("s_wait_asynccnt 0"  ::: "memory");
        __syncthreads();   // all waves' chunk-c data now in LDS

        const int buf = c & 1;
        const float* Ar0 = &Ab[buf][(wm * 2 + 0) * 16 + l15][khi];
        const float* Ar1 = &Ab[buf][(wm * 2 + 1) * 16 + l15][khi];
        const float* Br0 = &Wb[buf][(wn * 4 + 0) * 16 + l15][khi];
        const float* Br1 = &Wb[buf][(wn * 4 + 1) * 16 + l15][khi];
        const float* Br2 = &Wb[buf][(wn * 4 + 2) * 16 + l15][khi];
        const float* Br3 = &Wb[buf][(wn * 4 + 3) * 16 + l15][khi];

#pragma unroll
        for (int k = 0; k < KCH; k += 4) {
            v2f a0 = *(const v2f*)(Ar0 + k);
            v2f a1 = *(const v2f*)(Ar1 + k);
            v2f b0 = *(const v2f*)(Br0 + k);
            v2f b1 = *(const v2f*)(Br1 + k);
            v2f b2 = *(const v2f*)(Br2 + k);
            v2f b3 = *(const v2f*)(Br3 + k);
            acc[0][0] = __builtin_amdgcn_wmma_f32_16x16x4_f32(false, a0, false, b0, (short)0, acc[0][0], false, false);
            acc[0][1] = __builtin_amdgcn_wmma_f32_16x16x4_f32(false, a0, false, b1, (short)0, acc[0][1], false, false);
            acc[0][2] = __builtin_amdgcn_wmma_f32_16x16x4_f32(false, a0, false, b2, (short)0, acc[0][2], false, false);
            acc[0][3] = __builtin_amdgcn_wmma_f32_16x16x4_f32(false, a0, false, b3, (short)0, acc[0][3], false, false);
            acc[1][0] = __builtin_amdgcn_wmma_f32_16x16x4_f32(false, a1, false, b0, (short)0, acc[1][0], false, false);
            acc[1][1] = __builtin_amdgcn_wmma_f32_16x16x4_f32(false, a1, false, b1, (short)0, acc[1][1], false, false);
            acc[1][2] = __builtin_amdgcn_wmma_f32_16x16x4_f32(false, a1, false, b2, (short)0, acc[1][2], false, false);
            acc[1][3] = __builtin_amdgcn_wmma_f32_16x16x4_f32(false, a1, false, b3, (short)0, acc[1][3], false, false);
        }
        __syncthreads();   // everyone done reading buf before it is overwritten
        if (c + 2 < NCH) issue_chunk(c + 2);
    }

    // ---- Epilogue: bias + store. C/D layout: VGPR p -> M = p (lanes 0-15) / p+8 (16-31), N = lane%16 ----
    float bv[4];
#pragma unroll
    for (int nt = 0; nt < 4; ++nt) bv[nt] = bias[(wn * 4 + nt) * 16 + l15];

    const int hi = lane >> 4;
#pragma unroll
    for (int mt = 0; mt < 2; ++mt) {
        const int mbase = m0 + (wm * 2 + mt) * 16 + hi * 8;
#pragma unroll
        for (int nt = 0; nt < 4; ++nt) {
            const int col = (wn * 4 + nt) * 16 + l15;
#pragma unroll
            for (int p = 0; p < 8; ++p) {
                const int m = mbase + p;
                if (m < M_TOT) out[m * COUT + col] = acc[mt][nt][p] + bv[nt];
            }
        }
    }
}

extern "C" void kernel_launch(void* const* d_in, const int* in_sizes, int n_in,
                              void* d_out, int out_size, void* d_ws, size_t ws_size,
                              hipStream_t stream) {
    const float* x     = (const float*)d_in[0];
    const float* W     = (const float*)d_in[1];
    const float* b     = (const float*)d_in[2];
    const int*   neigh = (const int*)d_in[3];
    float*       ws    = (float*)d_ws;
    float*       out   = (float*)d_out;

    zero_ws_kernel<<<1, 64, 0, stream>>>(ws);

    const int grid = (M_TOT + M_TILE - 1) / M_TILE;   // 2561
    gconv_wmma_kernel<<<grid, 256, 0, stream>>>(x, W, b, neigh, ws, out);
}